// MatrixLSTMCell_81570018886143
// MI455X (gfx1250) — compile-verified
//
#include <hip/hip_runtime.h>

// MatrixLSTMCell on MI455X (gfx1250): fp32 WMMA 16x16x4 path.
// B=128, N=256 everywhere. Fused per-(batch, 16-row-block) workgroups.

#define N       256
#define BATCH   128
#define TSTRIDE 260   // 256 + 4 pad: 8-row offset = 2080 dw = +32 banks -> conflict-free

typedef __attribute__((ext_vector_type(2))) float v2f;
typedef __attribute__((ext_vector_type(8))) float v8f;

__device__ __forceinline__ v8f wmma4(v2f a, v2f b, v8f c) {
    // D = A(16x4) * B(4x16) + C(16x16), all fp32
    return __builtin_amdgcn_wmma_f32_16x16x4_f32(
        /*neg_a=*/false, a, /*neg_b=*/false, b,
        /*c_mod=*/(short)0, c, /*reuse_a=*/false, /*reuse_b=*/false);
}

// Compute 4 T-stripes (16 x 256 each): T_g = W_g[rowblock,:] @ X, into LDS.
// One B-operand load feeds 4 WMMAs (4 gate weights) -> X streamed once.
__device__ __forceinline__ void compute_T_stripes(
    const float* __restrict__ X,                      // [256][256] for this batch
    const float* __restrict__ w0, const float* __restrict__ w1,
    const float* __restrict__ w2, const float* __restrict__ w3,
    float* __restrict__ lds, int stripe0,             // 0 for x-source, 4 for h-source
    int row, int col0, int col1, int lhi, int lrow)
{
    const float* W[4] = {w0, w1, w2, w3};
    v8f acc[4][2] = {};
    for (int k = 0; k < N; k += 4) {
        const int kk = k + 2 * lhi;                   // A/B lane K split per ISA layout
        v2f b0, b1;
        b0.x = X[kk * N + col0]; b0.y = X[(kk + 1) * N + col0];
        b1.x = X[kk * N + col1]; b1.y = X[(kk + 1) * N + col1];
        #pragma unroll
        for (int g = 0; g < 4; ++g) {
            v2f a = *(const v2f*)(W[g] + row * N + kk);   // contiguous pair, b64
            acc[g][0] = wmma4(a, b0, acc[g][0]);
            acc[g][1] = wmma4(a, b1, acc[g][1]);
        }
    }
    // C/D layout: vgpr r, lanes0-15 -> (M=r, Nc=lane); lanes16-31 -> (M=r+8, Nc=lane-16)
    #pragma unroll
    for (int g = 0; g < 4; ++g) {
        #pragma unroll
        for (int r = 0; r < 8; ++r) {
            const int rr = lhi ? (r + 8) : r;
            lds[((stripe0 + g) * 16 + rr) * TSTRIDE + col0] = acc[g][0][r];
            lds[((stripe0 + g) * 16 + rr) * TSTRIDE + col1] = acc[g][1][r];
        }
    }
}

__global__ void __launch_bounds__(256)
gates_kernel(const float* __restrict__ input, const float* __restrict__ hidden,
             const float* __restrict__ cell,
             const float* __restrict__ Wfx, const float* __restrict__ Wix,
             const float* __restrict__ Wox, const float* __restrict__ Wgx,
             const float* __restrict__ Wfh, const float* __restrict__ Wih,
             const float* __restrict__ Woh, const float* __restrict__ Wgh,
             float* __restrict__ c_out, float* __restrict__ o_gate)
{
    extern __shared__ float lds[];        // 8 stripes x [16][TSTRIDE] = 130 KB
    const int rb   = blockIdx.x;          // 16-row block (0..15)
    const int b    = blockIdx.y;          // batch
    const int tid  = threadIdx.x;
    const int wave = tid >> 5;
    const int lane = tid & 31;
    const int lrow = lane & 15;
    const int lhi  = lane >> 4;
    const int row  = rb * 16 + lrow;      // A-matrix row (M = lane%16)
    const int col0 = (wave * 2) * 16 + lrow;   // each wave owns 2 of 16 N-tiles
    const int col1 = col0 + 16;

    const float* Xb = input  + (size_t)b * N * N;
    const float* Hb = hidden + (size_t)b * N * N;

    // Phase 1/2: T-stripes for x-source and h-source (stream each matrix once)
    compute_T_stripes(Xb, Wfx, Wix, Wox, Wgx, lds, 0, row, col0, col1, lhi, lrow);
    compute_T_stripes(Hb, Wfh, Wih, Woh, Wgh, lds, 4, row, col0, col1, lhi, lrow);
    __syncthreads();

    // Phase 3: S_g = T_gx @ W_gx^T + T_gh @ W_gh^T, relu-accumulate
    const float* Wx[4] = {Wfx, Wix, Wox, Wgx};
    const float* Wh[4] = {Wfh, Wih, Woh, Wgh};
    v8f c_acc[2] = {};
    v8f o_acc[2] = {};
    #pragma unroll
    for (int g = 0; g < 4; ++g) {
        v8f s0 = {}, s1 = {};
        const float* __restrict__ wx = Wx[g];
        const float* __restrict__ wh = Wh[g];
        const float* ldsx = lds + ( g      * 16 + lrow) * TSTRIDE;
        const float* ldsh = lds + ((4 + g) * 16 + lrow) * TSTRIDE;
        for (int k = 0; k < N; k += 4) {
            const int kk = k + 2 * lhi;
            v2f ax  = *(const v2f*)(ldsx + kk);           // ds_load_b64, bank-clean
            v2f ah  = *(const v2f*)(ldsh + kk);
            v2f bx0 = *(const v2f*)(wx + col0 * N + kk);  // B = W^T: B[k][n] = W[n][k]
            v2f bx1 = *(const v2f*)(wx + col1 * N + kk);
            v2f bh0 = *(const v2f*)(wh + col0 * N + kk);
            v2f bh1 = *(const v2f*)(wh + col1 * N + kk);
            s0 = wmma4(ax, bx0, s0);
            s1 = wmma4(ax, bx1, s1);
            s0 = wmma4(ah, bh0, s0);
            s1 = wmma4(ah, bh1, s1);
        }
        if (g == 2) {  // o-gate: stash (not part of c)
            #pragma unroll
            for (int r = 0; r < 8; ++r) {
                o_acc[0][r] = fmaxf(s0[r], 0.0f);
                o_acc[1][r] = fmaxf(s1[r], 0.0f);
            }
        } else {       // f, i, g gates: c += relu(S)
            #pragma unroll
            for (int r = 0; r < 8; ++r) {
                c_acc[0][r] += fmaxf(s0[r], 0.0f);
                c_acc[1][r] += fmaxf(s1[r], 0.0f);
            }
        }
    }

    // c = fgate + cell + igate + ggate ; write c and ogate stripes
    const float* Cb = cell   + (size_t)b * N * N;
    float*       cO = c_out  + (size_t)b * N * N;
    float*       oO = o_gate + (size_t)b * N * N;
    #pragma unroll
    for (int r = 0; r < 8; ++r) {
        const int ra = rb * 16 + (lhi ? (r + 8) : r);
        cO[ra * N + col0] = c_acc[0][r] + Cb[ra * N + col0];
        cO[ra * N + col1] = c_acc[1][r] + Cb[ra * N + col1];
        oO[ra * N + col0] = o_acc[0][r];
        oO[ra * N + col1] = o_acc[1][r];
    }
}

// h = ogate + relu(W_oc @ c @ W_oc^T)
__global__ void __launch_bounds__(256)
hout_kernel(const float* __restrict__ Woc,
            const float* __restrict__ cmat,
            const float* __restrict__ o_gate,
            float* __restrict__ h_out)
{
    __shared__ float lds[16 * TSTRIDE];
    const int rb   = blockIdx.x;
    const int b    = blockIdx.y;
    const int tid  = threadIdx.x;
    const int wave = tid >> 5;
    const int lane = tid & 31;
    const int lrow = lane & 15;
    const int lhi  = lane >> 4;
    const int row  = rb * 16 + lrow;
    const int col0 = (wave * 2) * 16 + lrow;
    const int col1 = col0 + 16;
    const float* Cb = cmat + (size_t)b * N * N;

    // T stripe = Woc[rowblock,:] @ c
    v8f t0 = {}, t1 = {};
    for (int k = 0; k < N; k += 4) {
        const int kk = k + 2 * lhi;
        v2f a = *(const v2f*)(Woc + row * N + kk);
        v2f b0, b1;
        b0.x = Cb[kk * N + col0]; b0.y = Cb[(kk + 1) * N + col0];
        b1.x = Cb[kk * N + col1]; b1.y = Cb[(kk + 1) * N + col1];
        t0 = wmma4(a, b0, t0);
        t1 = wmma4(a, b1, t1);
    }
    #pragma unroll
    for (int r = 0; r < 8; ++r) {
        const int rr = lhi ? (r + 8) : r;
        lds[rr * TSTRIDE + col0] = t0[r];
        lds[rr * TSTRIDE + col1] = t1[r];
    }
    __syncthreads();

    // H stripe = T @ Woc^T
    v8f s0 = {}, s1 = {};
    const float* ldsa = lds + lrow * TSTRIDE;
    for (int k = 0; k < N; k += 4) {
        const int kk = k + 2 * lhi;
        v2f a  = *(const v2f*)(ldsa + kk);
        v2f b0 = *(const v2f*)(Woc + col0 * N + kk);
        v2f b1 = *(const v2f*)(Woc + col1 * N + kk);
        s0 = wmma4(a, b0, s0);
        s1 = wmma4(a, b1, s1);
    }
    const float* oG = o_gate + (size_t)b * N * N;
    float*       hO = h_out  + (size_t)b * N * N;
    #pragma unroll
    for (int r = 0; r < 8; ++r) {
        const int ra = rb * 16 + (lhi ? (r + 8) : r);
        hO[ra * N + col0] = oG[ra * N + col0] + fmaxf(s0[r], 0.0f);
        hO[ra * N + col1] = oG[ra * N + col1] + fmaxf(s1[r], 0.0f);
    }
}

extern "C" void kernel_launch(void* const* d_in, const int* in_sizes, int n_in,
                              void* d_out, int out_size, void* d_ws, size_t ws_size,
                              hipStream_t stream) {
    (void)in_sizes; (void)n_in; (void)out_size; (void)ws_size;
    const float* input  = (const float*)d_in[0];
    const float* hidden = (const float*)d_in[1];
    const float* cell   = (const float*)d_in[2];
    const float* Wfx    = (const float*)d_in[3];
    const float* Wix    = (const float*)d_in[4];
    const float* Wox    = (const float*)d_in[5];
    const float* Wgx    = (const float*)d_in[6];
    const float* Wfh    = (const float*)d_in[7];
    const float* Wih    = (const float*)d_in[8];
    const float* Woh    = (const float*)d_in[9];
    const float* Wgh    = (const float*)d_in[10];
    const float* Woc    = (const float*)d_in[11];

    float* h_out = (float*)d_out;                          // tuple order: (h, c)
    float* c_out = h_out + (size_t)BATCH * N * N;
    float* o_ws  = (float*)d_ws;                           // 33.5 MB o-gate stash

    dim3 grid(N / 16, BATCH);                              // (row-block, batch)
    const size_t shmem = (size_t)8 * 16 * TSTRIDE * sizeof(float);  // 133,120 B

    gates_kernel<<<grid, 256, shmem, stream>>>(input, hidden, cell,
        Wfx, Wix, Wox, Wgx, Wfh, Wih, Woh, Wgh, c_out, o_ws);
    hout_kernel<<<grid, 256, 0, stream>>>(Woc, c_out, o_ws, h_out);
}